// GraphLearner_89137751261401
// MI455X (gfx1250) — compile-verified
//
#include <hip/hip_runtime.h>

// Problem constants (from the reference): B=2048, M=128, D=512, H=512.
#define BB   2048
#define MM   128
#define DD   512
#define HH   512

typedef float v2f __attribute__((ext_vector_type(2)));
typedef float v4f __attribute__((ext_vector_type(4)));
typedef float v8f __attribute__((ext_vector_type(8)));

// ---------------------------------------------------------------------------
// Kernel 1: streaming mean over the node axis M for both base feature tensors.
// Memory-bound: 1.07 GB streamed once -> wide non-temporal b128 loads,
// 4096 blocks x 4 waves to saturate 23.3 TB/s HBM.
// grid = (B, 2): y==0 -> text, y==1 -> image. block = 128 threads (v4f/thread).
// ---------------------------------------------------------------------------
__global__ void __launch_bounds__(128) mean_nodes_kernel(
    const float* __restrict__ base_txt, const float* __restrict__ base_img,
    float* __restrict__ txt_mean, float* __restrict__ img_mean)
{
    const int b = blockIdx.x;
    const float* __restrict__ src = blockIdx.y ? base_img : base_txt;
    float* __restrict__       dst = blockIdx.y ? img_mean : txt_mean;

    const int col = threadIdx.x * 4;                       // 128 threads * 4 = 512 = H
    const float* p = src + (size_t)b * (MM * HH) + col;

    v4f acc = {0.f, 0.f, 0.f, 0.f};
#pragma unroll 8
    for (int m = 0; m < MM; ++m) {
        v4f v = __builtin_nontemporal_load((const v4f*)(p + (size_t)m * HH));
        acc += v;
    }
    acc *= (1.0f / (float)MM);
    *(v4f*)(dst + (size_t)b * HH + col) = acc;
}

// ---------------------------------------------------------------------------
// WMMA fp32 helper: D(16x16) += A(16x4) * B(4x16), fp32 in / fp32 accumulate.
// Per the CDNA5 VGPR layout for V_WMMA_F32_16X16X4_F32:
//   lane L: m = n = L%16, koff = 2*(L/16); A vgpr j holds K = koff+j,
//   B vgpr j holds B[koff+j][n] = W[n][koff+j]; D vgpr r holds row r + 8*(L/16).
// Each wave accumulates FOUR adjacent 16x16 N-tiles: one A fetch feeds four
// independent WMMA chains (less A traffic, hides WMMA pipe latency).
// ---------------------------------------------------------------------------
__device__ __forceinline__ v8f wmma_f32(v2f a, v2f b, v8f c) {
    return __builtin_amdgcn_wmma_f32_16x16x4_f32(
        /*neg_a=*/false, a, /*neg_b=*/false, b,
        /*c_mod=*/(short)0, c, /*reuse_a=*/false, /*reuse_b=*/false);
}

// C = A[M x klen] (row stride lda) times W^T, W rows of stride ldw (row n holds
// the K-vector for output column n). Accumulates into c[0..3] (N-tiles j*16).
__device__ __forceinline__ void gemm_acc4(
    const float* __restrict__ A, int lda,
    const float* __restrict__ W, int ldw, int klen,
    int mrow, int ncol, int koff, v8f c[4])
{
    const float* Ar = A + (size_t)mrow * lda;
    const float* W0 = W + (size_t)(ncol +  0) * ldw;
    const float* W1 = W + (size_t)(ncol + 16) * ldw;
    const float* W2 = W + (size_t)(ncol + 32) * ldw;
    const float* W3 = W + (size_t)(ncol + 48) * ldw;
#pragma unroll 4
    for (int kb = 0; kb < klen; kb += 4) {
        const int k = kb + koff;                 // koff in {0,2}; 8B aligned
        v2f a  = *(const v2f*)(Ar + k);
        v2f b0 = *(const v2f*)(W0 + k);
        v2f b1 = *(const v2f*)(W1 + k);
        v2f b2 = *(const v2f*)(W2 + k);
        v2f b3 = *(const v2f*)(W3 + k);
        c[0] = wmma_f32(a, b0, c[0]);
        c[1] = wmma_f32(a, b1, c[1]);
        c[2] = wmma_f32(a, b2, c[2]);
        c[3] = wmma_f32(a, b3, c[3]);
    }
}

// ---------------------------------------------------------------------------
// Kernel 2: u = concat(it, ii) @ W_user^T + b_user      (2048 x 1024 x 512)
// K=1024 splits cleanly: k<512 reads input_text, k>=512 reads input_img
// (W_user row stride 1024, second half at +512).
// Wave job = one 16-row x 64-col output strip. 1024 jobs = 128 blocks x 8 waves.
// ---------------------------------------------------------------------------
__global__ void __launch_bounds__(256) user_gemm_kernel(
    const float* __restrict__ it, const float* __restrict__ ii,
    const float* __restrict__ Wu, const float* __restrict__ bu,
    float* __restrict__ u)
{
    const int lane = threadIdx.x & 31;
    const int wave = threadIdx.x >> 5;
    const int job  = blockIdx.x * 8 + wave;        // 0..1023
    const int tm   = job >> 3;                     // 0..127  (M tiles)
    const int ncol = (job & 7) * 64 + (lane & 15); // base col of 4-tile strip
    const int mrow = tm * 16 + (lane & 15);
    const int koff = (lane >> 4) * 2;

    v8f c[4] = {};
    gemm_acc4(it, DD, Wu,      2 * DD, DD, mrow, ncol, koff, c);  // k 0..511
    gemm_acc4(ii, DD, Wu + DD, 2 * DD, DD, mrow, ncol, koff, c);  // k 512..1023

    const int rbase = tm * 16 + (lane >> 4) * 8;
#pragma unroll
    for (int j = 0; j < 4; ++j) {
        const int col  = ncol + 16 * j;
        const float bv = bu[col];
#pragma unroll
        for (int r = 0; r < 8; ++r)
            u[(size_t)(rbase + r) * HH + col] = c[j][r] + bv;
    }
}

// ---------------------------------------------------------------------------
// Kernel 3: out = relu( u@W_r_img^T + u@W_r_txt^T
//                     + img_mean@W_l_img^T + txt_mean@W_l_txt^T
//                     + b_l_img + b_l_txt )
// Four K=512 accumulation passes into one fp32 WMMA accumulator set.
// ---------------------------------------------------------------------------
__global__ void __launch_bounds__(256) out_gemm_kernel(
    const float* __restrict__ u,
    const float* __restrict__ img_mean, const float* __restrict__ txt_mean,
    const float* __restrict__ Wli, const float* __restrict__ bli,
    const float* __restrict__ Wri,
    const float* __restrict__ Wlt, const float* __restrict__ blt,
    const float* __restrict__ Wrt,
    float* __restrict__ out)
{
    const int lane = threadIdx.x & 31;
    const int wave = threadIdx.x >> 5;
    const int job  = blockIdx.x * 8 + wave;        // 0..1023
    const int tm   = job >> 3;
    const int ncol = (job & 7) * 64 + (lane & 15);
    const int mrow = tm * 16 + (lane & 15);
    const int koff = (lane >> 4) * 2;

    v8f c[4] = {};
    gemm_acc4(u,        HH, Wri, HH, HH, mrow, ncol, koff, c);
    gemm_acc4(u,        HH, Wrt, HH, HH, mrow, ncol, koff, c);
    gemm_acc4(img_mean, HH, Wli, HH, HH, mrow, ncol, koff, c);
    gemm_acc4(txt_mean, HH, Wlt, HH, HH, mrow, ncol, koff, c);

    const int rbase = tm * 16 + (lane >> 4) * 8;
#pragma unroll
    for (int j = 0; j < 4; ++j) {
        const int col  = ncol + 16 * j;
        const float bv = bli[col] + blt[col];
#pragma unroll
        for (int r = 0; r < 8; ++r) {
            float v = c[j][r] + bv;
            out[(size_t)(rbase + r) * HH + col] = v > 0.f ? v : 0.f;
        }
    }
}

// ---------------------------------------------------------------------------
extern "C" void kernel_launch(void* const* d_in, const int* in_sizes, int n_in,
                              void* d_out, int out_size, void* d_ws, size_t ws_size,
                              hipStream_t stream)
{
    const float* input_text = (const float*)d_in[0];   // [B,1,D]
    const float* input_img  = (const float*)d_in[1];   // [B,1,D]
    const float* base_txt   = (const float*)d_in[2];   // [B,M,H]
    const float* base_img   = (const float*)d_in[3];   // [B,M,H]
    const float* W_user     = (const float*)d_in[4];   // [H, 2D]
    const float* b_user     = (const float*)d_in[5];   // [H]
    const float* W_l_img    = (const float*)d_in[6];   // [H, H]
    const float* b_l_img    = (const float*)d_in[7];   // [H]
    const float* W_r_img    = (const float*)d_in[8];   // [H, H]
    const float* W_l_txt    = (const float*)d_in[9];   // [H, H]
    const float* b_l_txt    = (const float*)d_in[10];  // [H]
    const float* W_r_txt    = (const float*)d_in[11];  // [H, H]
    float* out = (float*)d_out;

    // Workspace: u (4MB) | img_mean (4MB) | txt_mean (4MB)
    float* u        = (float*)d_ws;
    float* img_mean = u + (size_t)BB * HH;
    float* txt_mean = img_mean + (size_t)BB * HH;

    // 1) streaming means (dominant: ~1.07 GB @ HBM roofline)
    mean_nodes_kernel<<<dim3(BB, 2), 128, 0, stream>>>(
        base_txt, base_img, txt_mean, img_mean);

    // 2) u = concat(it, ii) @ W_user^T + b_user  (independent of the means)
    user_gemm_kernel<<<128, 256, 0, stream>>>(
        input_text, input_img, W_user, b_user, u);

    // 3) fused SAGE projections + biases + ReLU
    out_gemm_kernel<<<128, 256, 0, stream>>>(
        u, img_mean, txt_mean,
        W_l_img, b_l_img, W_r_img,
        W_l_txt, b_l_txt, W_r_txt,
        out);
}